// Decoder_49125835932141
// MI455X (gfx1250) — compile-verified
//
#include <hip/hip_runtime.h>
#include <math.h>

// ---------------- problem constants ----------------
constexpr int T_IN  = 24;
constexpr int BATCH = 8192;
constexpr int HID   = 512;
constexpr int T_OUT = 12;
constexpr int OVL   = 4;   // overlap
// 3*HID = 1536 gate width

typedef __attribute__((ext_vector_type(2))) float v2f;
typedef __attribute__((ext_vector_type(8))) float v8f;

// ---------------------------------------------------------------------------
// Init: copy encoder_hid -> hidden state, last -> last working buffer.
// Re-run every call so the launch is deterministic under graph replay.
// ---------------------------------------------------------------------------
__global__ __launch_bounds__(256) void init_state_kernel(
    const float* __restrict__ enc_hid,   // (B,H)
    const float* __restrict__ last_in,   // (B,4)
    float* __restrict__ hidden,          // (B,H)
    float* __restrict__ lastw)           // (B,4)
{
    size_t idx = (size_t)blockIdx.x * blockDim.x + threadIdx.x;
    size_t nh = (size_t)BATCH * HID;
    if (idx < nh) {
        hidden[idx] = enc_hid[idx];
    } else if (idx < nh + (size_t)BATCH * OVL) {
        size_t j = idx - nh;
        lastw[j] = last_in[j];
    }
}

// ---------------------------------------------------------------------------
// Attention: one workgroup per 4 batch elements. Stage the full (24,4,512)
// encoder slice in LDS (192KB + scratch) so encoder_out is read from HBM
// exactly ONCE per step (the bandwidth floor: 402MB/step @ 23.3TB/s).
// Computes context[b,:] and hatt = hidden + context.
// ---------------------------------------------------------------------------
__global__ __launch_bounds__(512) void attn_kernel(
    const float* __restrict__ enc,     // (24,B,H)
    const float* __restrict__ hidden,  // (B,H)
    float* __restrict__ ctx,           // (B,H)
    float* __restrict__ hatt)          // (B,H)
{
    extern __shared__ float smem[];
    float* tile  = smem;                       // 24*4*512 floats
    float* dvals = smem + T_IN * 4 * HID;      // 96 dot products
    float* wvals = dvals + T_IN * 4;           // 96 softmax weights

    const int b0  = blockIdx.x * 4;
    const int tid = threadIdx.x;               // 512 threads = 16 waves

    // ---- load tile: 96 rows x 512 floats, coalesced (tid spans one row) ----
    #pragma unroll 4
    for (int r = 0; r < T_IN * 4; ++r) {
        int t = r >> 2, bi = r & 3;
        tile[r * HID + tid] = enc[((size_t)t * BATCH + b0 + bi) * HID + tid];
    }
    __syncthreads();

    // ---- 96 dot products vs hidden rows; one wave32 per row (6 rows/wave) ----
    const int lane = tid & 31;
    const int wave = tid >> 5;
    for (int i = 0; i < 6; ++i) {
        int r  = wave + 16 * i;
        int bi = r & 3;
        const float* hrow = hidden + (size_t)(b0 + bi) * HID;
        float s = 0.f;
        #pragma unroll
        for (int q = lane; q < HID; q += 32)
            s += tile[r * HID + q] * hrow[q];
        #pragma unroll
        for (int m = 16; m >= 1; m >>= 1)
            s += __shfl_xor(s, m, 32);
        if (lane == 0) dvals[r] = s;
    }
    __syncthreads();

    // ---- softmax over t (24) per batch element: threads 0..3 ----
    if (tid < 4) {
        float mx = -1e30f;
        for (int t = 0; t < T_IN; ++t) mx = fmaxf(mx, dvals[t * 4 + tid]);
        float e[T_IN];
        float sum = 0.f;
        for (int t = 0; t < T_IN; ++t) {
            e[t] = __expf(dvals[t * 4 + tid] - mx);
            sum += e[t];
        }
        float inv = 1.f / sum;
        for (int t = 0; t < T_IN; ++t) wvals[t * 4 + tid] = e[t] * inv;
    }
    __syncthreads();

    // ---- context from the LDS-resident tile (no second HBM read) ----
    #pragma unroll
    for (int bi = 0; bi < 4; ++bi) {
        float c = 0.f;
        #pragma unroll
        for (int t = 0; t < T_IN; ++t)
            c += wvals[t * 4 + bi] * tile[(t * 4 + bi) * HID + tid];
        size_t o = (size_t)(b0 + bi) * HID + tid;
        ctx[o]  = c;
        hatt[o] = hidden[o] + c;
    }
}

// ---------------------------------------------------------------------------
// GRU cell with WMMA f32 16x16x4 for gh = hatt @ W_hh^T (12.9 GFLOP/step).
// WG = 128 threads (4 waves), WG tile = 64 rows x 64 gate-cols; each wave
// owns a 32x32 tile and accumulates all THREE gate blocks (r,z,n) so the
// full GRU epilogue (incl. the tiny K=8 W_ih matvec and nonlinearities)
// fuses here and the new hidden is written directly.
// Fragment mapping (ISA 7.12.2, 32-bit A 16x4):
//   A vgpr0/1 = A[m = base+(lane&15)][k0 + 2*(lane>>4) + {0,1}]  (float2 load)
//   B assumed symmetric: B[k0 + 2*(lane>>4) + {0,1}][n = base+(lane&15)]
//   C vgpr j : m = base + j + 8*(lane>>4), n = base + (lane&15)
// ---------------------------------------------------------------------------
__global__ __launch_bounds__(128) void gru_wmma_kernel(
    const float* __restrict__ hatt,   // (B,H)  = hidden + context
    const float* __restrict__ W_ih,   // (3H,8)
    const float* __restrict__ W_hh,   // (3H,H)
    const float* __restrict__ b_ih,   // (3H)
    const float* __restrict__ b_hh,   // (3H)
    const float* __restrict__ lastw,  // (B,4)
    const float* __restrict__ pos_embed, // (12,4)
    int step,
    float* __restrict__ hidden_out)   // (B,H)
{
    const int lane  = threadIdx.x & 31;
    const int wave  = threadIdx.x >> 5;
    const int waveM = wave >> 1;
    const int waveN = wave & 1;
    const int Mbase = blockIdx.x * 64 + waveM * 32;
    const int Nbase = blockIdx.y * 64 + waveN * 32;
    const int l15   = lane & 15;
    const int hi16  = lane >> 4;         // 0 or 1
    const int koff  = hi16 * 2;

    v8f acc[3][2][2];                    // [gate][sub_m][sub_n]
    #pragma unroll
    for (int g = 0; g < 3; ++g)
        #pragma unroll
        for (int sm = 0; sm < 2; ++sm)
            #pragma unroll
            for (int sn = 0; sn < 2; ++sn)
                acc[g][sm][sn] = (v8f){0.f, 0.f, 0.f, 0.f, 0.f, 0.f, 0.f, 0.f};

    // ---- K sweep: 128 iterations of K=4 f32 WMMA, 12 wmma per iteration ----
    for (int k = 0; k < HID; k += 4) {
        v2f afr[2];
        v2f bfr[3][2];
        #pragma unroll
        for (int sm = 0; sm < 2; ++sm) {
            int m = Mbase + sm * 16 + l15;
            afr[sm] = *(const v2f*)(hatt + (size_t)m * HID + k + koff);
        }
        #pragma unroll
        for (int g = 0; g < 3; ++g)
            #pragma unroll
            for (int sn = 0; sn < 2; ++sn) {
                int n = Nbase + sn * 16 + l15 + g * HID;
                bfr[g][sn] = *(const v2f*)(W_hh + (size_t)n * HID + k + koff);
            }
        #pragma unroll
        for (int g = 0; g < 3; ++g)
            #pragma unroll
            for (int sm = 0; sm < 2; ++sm)
                #pragma unroll
                for (int sn = 0; sn < 2; ++sn)
                    acc[g][sm][sn] = __builtin_amdgcn_wmma_f32_16x16x4_f32(
                        false, afr[sm], false, bfr[g][sn],
                        (short)0, acc[g][sm][sn], false, false);
    }

    // ---- fused GRU epilogue ----
    const float4 pe = *(const float4*)(pos_embed + step * 4);

    #pragma unroll
    for (int sm = 0; sm < 2; ++sm) {
        #pragma unroll
        for (int j = 0; j < 8; ++j) {
            const int m = Mbase + sm * 16 + j + 8 * hi16;
            const float4 xl = *(const float4*)(lastw + (size_t)m * OVL);
            const float x[8] = {xl.x, xl.y, xl.z, xl.w, pe.x, pe.y, pe.z, pe.w};
            #pragma unroll
            for (int sn = 0; sn < 2; ++sn) {
                const int n = Nbase + sn * 16 + l15;
                // gi = x @ W_ih^T + b_ih  (K=8, trivial VALU work)
                float gi[3];
                #pragma unroll
                for (int g = 0; g < 3; ++g) {
                    const float* wr = W_ih + (size_t)(g * HID + n) * 8;
                    float s = b_ih[g * HID + n];
                    #pragma unroll
                    for (int q = 0; q < 8; ++q) s += x[q] * wr[q];
                    gi[g] = s;
                }
                const float hr = acc[0][sm][sn][j] + b_hh[0 * HID + n];
                const float hz = acc[1][sm][sn][j] + b_hh[1 * HID + n];
                const float hn = acc[2][sm][sn][j] + b_hh[2 * HID + n];
                const float r  = 1.f / (1.f + __expf(-(gi[0] + hr)));
                const float z  = 1.f / (1.f + __expf(-(gi[1] + hz)));
                const float nn = tanhf(gi[2] + r * hn);
                const float hp = hatt[(size_t)m * HID + n];
                hidden_out[(size_t)m * HID + n] = (1.f - z) * nn + z * hp;
            }
        }
    }
}

// ---------------------------------------------------------------------------
// Output head: out[b] = [hidden | context] . W_lin + b_lin ; shift `last`.
// One wave32 per batch row.
// ---------------------------------------------------------------------------
__global__ __launch_bounds__(256) void out_kernel(
    const float* __restrict__ hidden,  // (B,H) (new)
    const float* __restrict__ ctx,     // (B,H)
    const float* __restrict__ W_lin,   // (1,2H)
    const float* __restrict__ b_lin,   // (1)
    float* __restrict__ lastw,         // (B,4) updated in place
    float* __restrict__ out,           // (B,12)
    int step)
{
    const int wave = threadIdx.x >> 5;
    const int lane = threadIdx.x & 31;
    const int b = blockIdx.x * 8 + wave;
    const float* hr = hidden + (size_t)b * HID;
    const float* cr = ctx + (size_t)b * HID;
    float s = 0.f;
    #pragma unroll
    for (int q = lane; q < HID; q += 32)
        s += hr[q] * W_lin[q] + cr[q] * W_lin[HID + q];
    #pragma unroll
    for (int m = 16; m >= 1; m >>= 1)
        s += __shfl_xor(s, m, 32);
    if (lane == 0) {
        const float o = s + b_lin[0];
        out[(size_t)b * T_OUT + step] = o;
        const float l0 = lastw[b * OVL + 0];
        const float l1 = lastw[b * OVL + 1];
        const float l2 = lastw[b * OVL + 2];
        lastw[b * OVL + 0] = o;
        lastw[b * OVL + 1] = l0;
        lastw[b * OVL + 2] = l1;
        lastw[b * OVL + 3] = l2;
    }
}

// ---------------------------------------------------------------------------
extern "C" void kernel_launch(void* const* d_in, const int* in_sizes, int n_in,
                              void* d_out, int out_size, void* d_ws, size_t ws_size,
                              hipStream_t stream) {
    (void)in_sizes; (void)n_in; (void)out_size; (void)ws_size;
    const float* enc     = (const float*)d_in[0];  // (24,B,H)
    const float* enc_hid = (const float*)d_in[1];  // (B,H)
    const float* last_in = (const float*)d_in[2];  // (B,4)
    const float* pe      = (const float*)d_in[3];  // (12,4)
    const float* W_ih    = (const float*)d_in[4];  // (3H,8)
    const float* W_hh    = (const float*)d_in[5];  // (3H,H)
    const float* b_ih    = (const float*)d_in[6];  // (3H)
    const float* b_hh    = (const float*)d_in[7];  // (3H)
    const float* W_lin   = (const float*)d_in[8];  // (1,2H)
    const float* b_lin   = (const float*)d_in[9];  // (1)
    float* out = (float*)d_out;                    // (B,12)

    // workspace: hidden | context | hatt | last   (~48.1 MB)
    float* hidden = (float*)d_ws;
    float* ctx    = hidden + (size_t)BATCH * HID;
    float* hatt   = ctx    + (size_t)BATCH * HID;
    float* lastw  = hatt   + (size_t)BATCH * HID;

    {
        size_t n = (size_t)BATCH * HID + (size_t)BATCH * OVL;
        init_state_kernel<<<(unsigned)((n + 255) / 256), 256, 0, stream>>>(
            enc_hid, last_in, hidden, lastw);
    }

    const size_t attn_lds = (size_t)(T_IN * 4 * HID + 2 * T_IN * 4) * sizeof(float);

    for (int step = 0; step < T_OUT; ++step) {
        attn_kernel<<<BATCH / 4, 512, attn_lds, stream>>>(enc, hidden, ctx, hatt);
        gru_wmma_kernel<<<dim3(BATCH / 64, HID / 64), 128, 0, stream>>>(
            hatt, W_ih, W_hh, b_ih, b_hh, lastw, pe, step, hidden);
        out_kernel<<<BATCH / 8, 256, 0, stream>>>(
            hidden, ctx, W_lin, b_lin, lastw, out, step);
    }
}